// FeaturesMap_3624952398102
// MI455X (gfx1250) — compile-verified
//
#include <hip/hip_runtime.h>
#include <stdint.h>

#define BB 64
#define CC 512
#define NN 3000
#define MAXH 70
#define MAXW 40
#define NPIX (MAXH * MAXW)   // 2800

typedef float v4f __attribute__((ext_vector_type(4)));
typedef int   v4i __attribute__((ext_vector_type(4)));

// ---------------------------------------------------------------------------
// Kernel 1: per-batch masked bbox + crop/pad parameter computation.
// params[b*8 + {0..4}] = min_y, min_x, off_r, off_c, swap
// ---------------------------------------------------------------------------
__global__ void bbox_kernel(const int* __restrict__ ys,
                            const int* __restrict__ xs,
                            int* __restrict__ params) {
    __shared__ int sm0[128], sm1[128], sm2[128], sm3[128];
    const int b = blockIdx.x;
    const int t = threadIdx.x;
    int mny = 0x7fffffff, mnx = 0x7fffffff;
    int mxy = -0x7fffffff, mxx = -0x7fffffff;
    const int base = b * NN;
    for (int n = t; n < NN; n += 128) {
        int y = ys[base + n];
        int x = xs[base + n];
        if (y > -1) {
            mny = min(mny, y); mxy = max(mxy, y);
            mnx = min(mnx, x); mxx = max(mxx, x);
        }
    }
    sm0[t] = mny; sm1[t] = mnx; sm2[t] = mxy; sm3[t] = mxx;
    __syncthreads();
    for (int s = 64; s > 0; s >>= 1) {
        if (t < s) {
            sm0[t] = min(sm0[t], sm0[t + s]);
            sm1[t] = min(sm1[t], sm1[t + s]);
            sm2[t] = max(sm2[t], sm2[t + s]);
            sm3[t] = max(sm3[t], sm3[t + s]);
        }
        __syncthreads();
    }
    if (t == 0) {
        int min_y = sm0[0], min_x = sm1[0];
        int max_y = sm2[0], max_x = sm3[0];
        int h = max_y - min_y + 1;
        int w = max_x - min_x + 1;
        int swapf = (w > h) ? 1 : 0;
        int H0 = swapf ? w : h;
        int W0 = swapf ? h : w;
        int h_dif = H0 - MAXH;
        int w_dif = W0 - MAXW;
        int cut_top  = (h_dif > 0) ? (h_dif + 1) / 2 : 0;
        int pad_top  = (h_dif > 0) ? 0 : (1 - h_dif) / 2;
        int cut_left = (w_dif > 0) ? (w_dif + 1) / 2 : 0;
        int left_pad = (w_dif > 0) ? 0 : (1 - w_dif) / 2;
        int* pp = params + b * 8;
        pp[0] = min_y;
        pp[1] = min_x;
        pp[2] = pad_top - cut_top;    // off_r
        pp[3] = left_pad - cut_left;  // off_c
        pp[4] = swapf;
    }
}

// ---------------------------------------------------------------------------
// Kernel 2: build inverse map idx_map[b, rr*MAXW+cc] = n (targets unique).
// idx_map pre-initialized to -1 by hipMemsetAsync(0xFF).
// ---------------------------------------------------------------------------
__global__ void build_map_kernel(const int* __restrict__ ys,
                                 const int* __restrict__ xs,
                                 const int* __restrict__ params,
                                 int* __restrict__ idx_map) {
    int i = blockIdx.x * 256 + threadIdx.x;
    if (i >= BB * NN) return;
    int b = i / NN;
    int n = i - b * NN;
    int y = ys[i];
    if (y <= -1) return;
    int x = xs[i];
    const int* pp = params + b * 8;
    int min_y = pp[0], min_x = pp[1], off_r = pp[2], off_c = pp[3], swapf = pp[4];
    int dy = y - min_y;
    int dx = x - min_x;
    int r = swapf ? dx : dy;
    int c = swapf ? dy : dx;
    int rr = r + off_r;
    int cc = c + off_c;
    if (rr >= 0 && rr < MAXH && cc >= 0 && cc < MAXW) {
        idx_map[b * NPIX + rr * MAXW + cc] = n;
    }
}

// ---------------------------------------------------------------------------
// Kernel 3: gather. One block per (b,c) plane. Stage the 12 KB feature row in
// LDS via gfx1250 async-to-LDS (B128 per lane, ASYNCcnt, NT hint: features are
// read exactly once — keep L2 for the high-reuse idx_map), then emit coalesced
// non-temporal float4 stores of the 2800-pixel output plane. Flag any pixel
// that received an exact -1.0f (the reference's real2d test).
// ---------------------------------------------------------------------------
__global__ void gather_kernel(const float* __restrict__ features,
                              const int* __restrict__ idx_map,
                              float* __restrict__ out,
                              int* __restrict__ flags) {
    __shared__ float frow[NN];
    const int bc = blockIdx.x;           // 0 .. B*C-1
    const int b  = bc >> 9;              // / 512
    const int t  = threadIdx.x;          // 256 threads (8 waves)

    const float* fr = features + (size_t)bc * NN;
    uint32_t lds_base = (uint32_t)(size_t)(void*)&frow[0];

    // 3000 floats = 750 x 16B chunks; 3 rounds x 256 lanes (EXEC-masked tail)
    #pragma unroll
    for (int r = 0; r < 3; ++r) {
        int chunk = r * 256 + t;
        if (chunk < NN / 4) {
            uint32_t lds_off = lds_base + (uint32_t)chunk * 16u;
            const float* g = fr + chunk * 4;
            asm volatile("global_load_async_to_lds_b128 %0, %1, off th:TH_LOAD_NT"
                         :: "v"(lds_off), "v"(g) : "memory");
        }
    }
    asm volatile("s_wait_asynccnt 0" ::: "memory");
    __syncthreads();

    const int* imap = idx_map + b * NPIX;
    float*     orow = out + (size_t)bc * NPIX;
    int*       flg  = flags + b * NPIX;

    // 2800 pixels = 700 float4 chunks; 256 threads -> 3 strided passes
    for (int chunk = t; chunk < NPIX / 4; chunk += 256) {
        const int p0 = chunk * 4;
        v4i idx4 = *(const v4i*)(imap + p0);   // 16B-aligned B128 load
        v4f v;
        #pragma unroll
        for (int k = 0; k < 4; ++k) {
            int idx = idx4[k];
            float f = 0.0f;
            if (idx >= 0) {
                f = frow[idx];
                if (f == -1.0f) flg[p0 + k] = 1;   // benign race: all write 1
            }
            v[k] = f;
        }
        __builtin_nontemporal_store(v, (v4f*)(orow) + chunk);
    }
}

// ---------------------------------------------------------------------------
// Kernel 4: fixup — zero all channels of flagged pixels (expected ~3 total).
// ---------------------------------------------------------------------------
__global__ void fixup_kernel(const int* __restrict__ flags,
                             float* __restrict__ out) {
    int i = blockIdx.x * 256 + threadIdx.x;   // over B*NPIX
    if (i >= BB * NPIX) return;
    if (flags[i]) {
        int b = i / NPIX;
        int p = i - b * NPIX;
        float* o = out + (size_t)b * CC * NPIX + p;
        for (int c = 0; c < CC; ++c) {
            o[(size_t)c * NPIX] = 0.0f;
        }
    }
}

// ---------------------------------------------------------------------------
extern "C" void kernel_launch(void* const* d_in, const int* in_sizes, int n_in,
                              void* d_out, int out_size, void* d_ws, size_t ws_size,
                              hipStream_t stream) {
    (void)in_sizes; (void)n_in; (void)out_size; (void)ws_size;

    const float* features = (const float*)d_in[0];
    const int*   ys       = (const int*)d_in[1];
    const int*   xs       = (const int*)d_in[2];
    float*       out      = (float*)d_out;

    char* ws = (char*)d_ws;
    int* params  = (int*)(ws);                 // 64*8*4      = 2 KB
    int* idx_map = (int*)(ws + 4096);          // 64*2800*4   = 700 KB
    int* flags   = (int*)(ws + 4096 + BB * NPIX * 4);

    // Per-call re-init of scratch (graph-capturable async memsets).
    (void)hipMemsetAsync(idx_map, 0xFF, (size_t)BB * NPIX * 4, stream);  // -1
    (void)hipMemsetAsync(flags,   0x00, (size_t)BB * NPIX * 4, stream);

    bbox_kernel<<<BB, 128, 0, stream>>>(ys, xs, params);

    build_map_kernel<<<(BB * NN + 255) / 256, 256, 0, stream>>>(
        ys, xs, params, idx_map);

    gather_kernel<<<BB * CC, 256, 0, stream>>>(features, idx_map, out, flags);

    fixup_kernel<<<(BB * NPIX + 255) / 256, 256, 0, stream>>>(flags, out);
}